// SEGT_7464653161212
// MI455X (gfx1250) — compile-verified
//
#include <hip/hip_runtime.h>
#include <hip/hip_bf16.h>

typedef __attribute__((ext_vector_type(2))) float v2f;
typedef __attribute__((ext_vector_type(8))) float v8f;

#define BATCH 8
#define CCH   64
#define PIX   1024
#define PQ    256
#define GRP   32
#define HW    16384
#define KTOP  21

// ---------------------------------------------------------------------------
// 4x4 average pooling: x[B,64,128,128] -> pooled[B,64,1024]
__global__ void pool_kernel(const float* __restrict__ x, float* __restrict__ pooled, int total) {
    int idx = blockIdx.x * blockDim.x + threadIdx.x;
    if (idx >= total) return;
    int p  = idx & (PIX - 1);
    int bc = idx >> 10;
    int ph = p >> 5, pw = p & 31;
    const float* base = x + (size_t)bc * HW + (ph * 4) * 128 + pw * 4;
    float s = 0.f;
#pragma unroll
    for (int i = 0; i < 4; ++i)
#pragma unroll
        for (int j = 0; j < 4; ++j) s += base[i * 128 + j];
    pooled[idx] = s * (1.f / 16.f);
}

// xpre[b,g,p] = mean over channel pair (2g, 2g+1)
__global__ void xpre_kernel(const float* __restrict__ pooled, float* __restrict__ xpre, int total) {
    int idx = blockIdx.x * blockDim.x + threadIdx.x;
    if (idx >= total) return;
    int p = idx & (PIX - 1);
    int r = idx >> 10;          // b*32+g
    int b = r >> 5;
    int g = r & 31;
    const float* pb = pooled + ((size_t)(b * CCH + 2 * g)) * PIX + p;
    xpre[idx] = 0.5f * (pb[0] + pb[PIX]);
}

// ---------------------------------------------------------------------------
// WMMA projection: out[b,o,t] = dot(W[(b),o,:K], X[b,t,:K]) + bias[o]
// T = NT*16 columns; one wave per (b, 16-row tile of O), NT accumulators.
// A tile: lane(row=lane&15) holds W[m0+row][ka..ka+1], halves split K pairs.
// B tile: lane holds X[nt*16+row][ka..ka+1] (both are aligned b64 loads).
template <int NT>
__global__ void wmma_proj_kernel(const float* __restrict__ W, const float* __restrict__ X,
                                 const float* __restrict__ bias, float* __restrict__ out,
                                 int O, int K, long wBatchStride, long xBatchStride,
                                 int outTrans, int nwaves) {
    int wid  = (blockIdx.x * blockDim.x + threadIdx.x) >> 5;
    int lane = threadIdx.x & 31;
    if (wid >= nwaves) return;
    int mtiles = O >> 4;
    int b  = wid / mtiles;
    int m0 = (wid - b * mtiles) << 4;
    const float* Wb = W + (size_t)b * wBatchStride + (size_t)m0 * K;
    const float* Xb = X + (size_t)b * xBatchStride;
    const int T = NT * 16;
    int row = lane & 15, hi = lane >> 4;
    v8f z = {};
    v8f c[NT];
#pragma unroll
    for (int i = 0; i < NT; ++i) c[i] = z;
    for (int k4 = 0; k4 < K; k4 += 4) {
        int ka = k4 + 2 * hi;
        v2f a = *(const v2f*)(Wb + (size_t)row * K + ka);
#pragma unroll
        for (int nt = 0; nt < NT; ++nt) {
            v2f bb = *(const v2f*)(Xb + (size_t)(nt * 16 + row) * K + ka);
            c[nt] = __builtin_amdgcn_wmma_f32_16x16x4_f32(false, a, false, bb,
                                                          (short)0, c[nt], false, false);
        }
    }
    float* ob = out + (size_t)b * O * T;
#pragma unroll
    for (int nt = 0; nt < NT; ++nt) {
#pragma unroll
        for (int r = 0; r < 8; ++r) {
            int orow = m0 + r + 8 * hi;
            int ocol = nt * 16 + row;
            float v = c[nt][r] + (bias ? bias[orow] : 0.f);
            if (outTrans) ob[(size_t)ocol * O + orow] = v;
            else          ob[(size_t)orow * T + ocol] = v;
        }
    }
}

// WMMA row-major GEMM: out[b][M,N] = A[b][M,K] @ Bm[(b)][K,N] + bias[n]
// M = MT*16; one wave per (b, 16-col tile of N), MT accumulators.
template <int MT>
__global__ void wmma_gemm_nn_kernel(const float* __restrict__ A, const float* __restrict__ Bm,
                                    const float* __restrict__ bias, float* __restrict__ out,
                                    int N, int K, long aBatchStride, long bBatchStride,
                                    int nwaves) {
    int wid  = (blockIdx.x * blockDim.x + threadIdx.x) >> 5;
    int lane = threadIdx.x & 31;
    if (wid >= nwaves) return;
    int ntiles = N >> 4;
    int b  = wid / ntiles;
    int n0 = (wid - b * ntiles) << 4;
    const float* Ab = A + (size_t)b * aBatchStride;
    const float* Bb = Bm + (size_t)b * bBatchStride;
    const int M = MT * 16;
    int row = lane & 15, hi = lane >> 4;
    v8f z = {};
    v8f c[MT];
#pragma unroll
    for (int i = 0; i < MT; ++i) c[i] = z;
    for (int k4 = 0; k4 < K; k4 += 4) {
        int ka = k4 + 2 * hi;
        v2f bb;
        bb.x = Bb[(size_t)ka * N + n0 + row];
        bb.y = Bb[(size_t)(ka + 1) * N + n0 + row];
#pragma unroll
        for (int mt = 0; mt < MT; ++mt) {
            v2f a = *(const v2f*)(Ab + (size_t)(mt * 16 + row) * K + ka);
            c[mt] = __builtin_amdgcn_wmma_f32_16x16x4_f32(false, a, false, bb,
                                                          (short)0, c[mt], false, false);
        }
    }
    float* ob = out + (size_t)b * M * N;
    float bv = bias ? bias[n0 + row] : 0.f;
#pragma unroll
    for (int mt = 0; mt < MT; ++mt)
#pragma unroll
        for (int r = 0; r < 8; ++r)
            ob[(size_t)(mt * 16 + r + 8 * hi) * N + n0 + row] = c[mt][r] + bv;
}

// ---------------------------------------------------------------------------
// adj logits (32x32) + softmax + top-21 mask + re-softmax; one wave per (b,s)
__global__ void adj_kernel(const float* __restrict__ kT, const float* __restrict__ qg,
                           float* __restrict__ adj, int nwaves) {
    int wid  = (blockIdx.x * blockDim.x + threadIdx.x) >> 5;
    int lane = threadIdx.x & 31;
    if (wid >= nwaves) return;
    int b = wid >> 5;
    int s = wid & 31;
    const float* krow = kT + ((size_t)(b * GRP + s)) * PQ;
    const float* q    = qg + (size_t)b * PQ * GRP;
    float acc = 0.f;
    for (int d = 0; d < PQ; ++d) acc += krow[d] * q[d * GRP + lane];
    float m = acc;
    for (int off = 16; off; off >>= 1) m = fmaxf(m, __shfl_xor(m, off, 32));
    float e = __expf(acc - m);
    float ssum = e;
    for (int off = 16; off; off >>= 1) ssum += __shfl_xor(ssum, off, 32);
    float p = e / ssum;
    int rank = 0;
    for (int j = 0; j < 32; ++j) {
        float pj = __shfl(p, j, 32);
        rank += (pj > p) || (pj == p && j < lane);
    }
    bool keep = rank < KTOP;
    float v = keep ? p : -1e30f;
    float m2 = v;
    for (int off = 16; off; off >>= 1) m2 = fmaxf(m2, __shfl_xor(m2, off, 32));
    float e2 = keep ? __expf(p - m2) : 0.f;
    float s2 = e2;
    for (int off = 16; off; off >>= 1) s2 += __shfl_xor(s2, off, 32);
    adj[((size_t)(b * GRP + s)) * GRP + lane] = e2 / s2;
}

// attention logits (64x32) + softmax over 32; one wave per (b,c)
__global__ void atten_kernel(const float* __restrict__ key, const float* __restrict__ query,
                             float* __restrict__ atten, int nwaves) {
    int wid  = (blockIdx.x * blockDim.x + threadIdx.x) >> 5;
    int lane = threadIdx.x & 31;
    if (wid >= nwaves) return;
    int b = wid >> 6;
    int c = wid & 63;
    const float* krow = key + ((size_t)(b * CCH + c)) * PQ;
    const float* q    = query + (size_t)b * PQ * GRP;
    float acc = 0.f;
    for (int d = 0; d < PQ; ++d) acc += krow[d] * q[d * GRP + lane];
    float m = acc;
    for (int off = 16; off; off >>= 1) m = fmaxf(m, __shfl_xor(m, off, 32));
    float e = __expf(acc - m);
    float ssum = e;
    for (int off = 16; off; off >>= 1) ssum += __shfl_xor(ssum, off, 32);
    atten[((size_t)(b * CCH + c)) * GRP + lane] = e / ssum;
}

// tu2 flat reinterpretation: tu2_flat[b,i] = tu[b, i%64, i/64]
__global__ void tu2_kernel(const float* __restrict__ tu, float* __restrict__ tu2, int total) {
    int idx = blockIdx.x * blockDim.x + threadIdx.x;
    if (idx >= total) return;
    int b = idx >> 16;
    int i = idx & 65535;
    tu2[((size_t)b << 16) + i] = tu[((size_t)b << 16) + (size_t)(i & 63) * PIX + (i >> 6)];
}

// ---------------------------------------------------------------------------
// out[b] = Mmat[b] (64x64) @ x[b] (64x16384), fp32 WMMA.
// One wave owns a full 64-row x 16-col strip => x is streamed exactly once.
// M (16KB) is staged in LDS per block; all 8 waves of a block share one batch.
__global__ void out_wmma_kernel(const float* __restrict__ Mmat, const float* __restrict__ X,
                                float* __restrict__ out) {
    __shared__ float lM[CCH * CCH];
    int blk  = blockIdx.x;            // 1024 blocks: 128 per batch
    int b    = blk >> 7;
    int nblk = blk & 127;
    int wave = threadIdx.x >> 5;
    int lane = threadIdx.x & 31;
    int n0   = ((nblk << 3) + wave) << 4;      // 16-col strip for this wave
    const float* Mb = Mmat + ((size_t)b << 12);
    for (int i = threadIdx.x; i < CCH * CCH; i += 256) lM[i] = Mb[i];
    __syncthreads();
    const float* Xb = X + ((size_t)b * CCH) * HW;
    float* Ob       = out + ((size_t)b * CCH) * HW;
    int row = lane & 15, hi = lane >> 4;
    v8f z = {};
    v8f c[4] = {z, z, z, z};
    for (int k4 = 0; k4 < 64; k4 += 4) {
        int ka = k4 + 2 * hi;
        v2f bb;
        bb.x = Xb[(size_t)ka * HW + n0 + row];
        bb.y = Xb[(size_t)(ka + 1) * HW + n0 + row];
#pragma unroll
        for (int mt = 0; mt < 4; ++mt) {
            v2f a = *(const v2f*)(&lM[(mt * 16 + row) * 64 + ka]);
            c[mt] = __builtin_amdgcn_wmma_f32_16x16x4_f32(false, a, false, bb,
                                                          (short)0, c[mt], false, false);
        }
    }
#pragma unroll
    for (int mt = 0; mt < 4; ++mt)
#pragma unroll
        for (int r = 0; r < 8; ++r)
            Ob[(size_t)(mt * 16 + r + 8 * hi) * HW + n0 + row] = c[mt][r];
}

// ---------------------------------------------------------------------------
static inline int gridFor(int n, int blk) { return (n + blk - 1) / blk; }

extern "C" void kernel_launch(void* const* d_in, const int* in_sizes, int n_in,
                              void* d_out, int out_size, void* d_ws, size_t ws_size,
                              hipStream_t stream) {
    const float* x        = (const float*)d_in[0];
    const float* gcn_w    = (const float*)d_in[1];
    const float* gcn_b    = (const float*)d_in[2];
    const float* gcn_q_w  = (const float*)d_in[3];
    const float* gcn_q_b  = (const float*)d_in[4];
    const float* gcn_k_w  = (const float*)d_in[5];
    const float* gcn_k_b  = (const float*)d_in[6];
    const float* gcn_v_w  = (const float*)d_in[7];
    const float* gcn_v_b  = (const float*)d_in[8];
    const float* q_w      = (const float*)d_in[9];
    const float* q_b      = (const float*)d_in[10];
    const float* k_w      = (const float*)d_in[11];
    const float* k_b      = (const float*)d_in[12];
    const float* v_w      = (const float*)d_in[13];
    const float* v_b      = (const float*)d_in[14];
    float* out = (float*)d_out;

    float* ws = (float*)d_ws;
    float* pooled  = ws;  ws += BATCH * CCH * PIX;
    float* xpre    = ws;  ws += BATCH * GRP * PIX;
    float* qg      = ws;  ws += BATCH * PQ * GRP;
    float* kTg     = ws;  ws += BATCH * GRP * PQ;
    float* adj     = ws;  ws += BATCH * GRP * GRP;
    float* vTg     = ws;  ws += BATCH * GRP * PIX;
    float* support = ws;  ws += BATCH * GRP * PIX;
    float* gout    = ws;  ws += BATCH * GRP * PIX;
    float* query   = ws;  ws += BATCH * PQ * GRP;
    float* keyb    = ws;  ws += BATCH * CCH * PQ;
    float* atten   = ws;  ws += BATCH * CCH * GRP;
    float* value   = ws;  ws += BATCH * GRP * PIX;
    float* tu      = ws;  ws += BATCH * CCH * PIX;
    float* tu2     = ws;  ws += BATCH * CCH * PIX;
    float* Mmat    = ws;  ws += BATCH * CCH * CCH;

    const int BLK = 256;
    int n, w;

    // 1) pooling + group mean
    n = BATCH * CCH * PIX;
    pool_kernel<<<gridFor(n, BLK), BLK, 0, stream>>>(x, pooled, n);
    n = BATCH * GRP * PIX;
    xpre_kernel<<<gridFor(n, BLK), BLK, 0, stream>>>(pooled, xpre, n);

    // 2) GCN projections (WMMA): q [256,32], kT [32,256], vT [32,1024]
    w = BATCH * (PQ / 16);   // 128 waves
    wmma_proj_kernel<2><<<w / 8, BLK, 0, stream>>>(gcn_q_w, xpre, gcn_q_b, qg,
                                                   PQ, PIX, 0, GRP * PIX, 0, w);
    wmma_proj_kernel<2><<<w / 8, BLK, 0, stream>>>(gcn_k_w, xpre, gcn_k_b, kTg,
                                                   PQ, PIX, 0, GRP * PIX, 1, w);
    w = BATCH * (PIX / 16);  // 512 waves
    wmma_proj_kernel<2><<<w / 8, BLK, 0, stream>>>(gcn_v_w, xpre, gcn_v_b, vTg,
                                                   PIX, PIX, 0, GRP * PIX, 1, w);

    // 3) adjacency: logits + softmax + top-21 + re-softmax
    adj_kernel<<<gridFor(BATCH * GRP * 32, BLK), BLK, 0, stream>>>(kTg, qg, adj, BATCH * GRP);

    // 4) support = vT @ gcn_weight ; gout = adj @ support + gcn_bias  (WMMA)
    w = BATCH * (PIX / 16);  // 512 waves
    wmma_gemm_nn_kernel<2><<<w / 8, BLK, 0, stream>>>(vTg, gcn_w, nullptr, support,
                                                      PIX, PIX, GRP * PIX, 0, w);
    wmma_gemm_nn_kernel<2><<<w / 8, BLK, 0, stream>>>(adj, support, gcn_b, gout,
                                                      PIX, GRP, GRP * GRP, GRP * PIX, w);

    // 5) attention projections (WMMA)
    w = BATCH * (PQ / 16);
    wmma_proj_kernel<2><<<w / 8, BLK, 0, stream>>>(q_w, gout, q_b, query,
                                                   PQ, PIX, 0, GRP * PIX, 0, w);
    wmma_proj_kernel<4><<<w / 8, BLK, 0, stream>>>(k_w, pooled, k_b, keyb,
                                                   PQ, PIX, 0, CCH * PIX, 1, w);
    w = BATCH * (PIX / 16);
    wmma_proj_kernel<2><<<w / 8, BLK, 0, stream>>>(v_w, gout, v_b, value,
                                                   PIX, PIX, 0, GRP * PIX, 1, w);

    // 6) atten softmax
    atten_kernel<<<gridFor(BATCH * CCH * 32, BLK), BLK, 0, stream>>>(keyb, query, atten,
                                                                     BATCH * CCH);

    // 7) tu = atten @ value (WMMA)
    w = BATCH * (PIX / 16);
    wmma_gemm_nn_kernel<4><<<w / 8, BLK, 0, stream>>>(atten, value, nullptr, tu,
                                                      PIX, GRP, CCH * GRP, GRP * PIX, w);

    // 8) tu2 permutation + Gram matrix M = tu2 @ tu2^T (WMMA, per-batch W)
    n = BATCH * CCH * PIX;
    tu2_kernel<<<gridFor(n, BLK), BLK, 0, stream>>>(tu, tu2, n);
    w = BATCH * (CCH / 16);  // 32 waves
    wmma_proj_kernel<4><<<w / 8, BLK, 0, stream>>>(tu2, tu2, nullptr, Mmat,
                                                   CCH, PIX, CCH * PIX, CCH * PIX, 0, w);

    // 9) out = M @ x : bandwidth-bound tail, x streamed exactly once.
    //    1024 blocks x 8 waves; each wave = full 64x16 strip; M staged in LDS.
    out_wmma_kernel<<<BATCH * 128, BLK, 0, stream>>>(Mmat, x, out);
}